// GPFullLoss_10213432230328
// MI455X (gfx1250) — compile-verified
//
#include <hip/hip_runtime.h>
#include <hip/hip_bf16.h>
#include <math.h>

// ---------------------------------------------------------------------------
// GPFullLoss for MI455X (gfx1250, wave32)
// pred/gt: [16, 68, 256, 256] f32.  Output: scalar f32.
// Phase 1: channel-sum streaming reduction (HBM-bound, ~570MB -> ~24.5us floor)
//          float4 non-temporal loads; RT stores keep p/g (8MB) in 192MB L2.
// Phase 2: per-row/col cosine terms via V_WMMA_F32_16X16X4_F32 Gram diagonals.
// Phase 3: deterministic single-block finalize.
// ---------------------------------------------------------------------------

typedef float f4  __attribute__((ext_vector_type(4)));
typedef float v2f __attribute__((ext_vector_type(2)));
typedef float v8f __attribute__((ext_vector_type(8)));

#define NB   16
#define NC   68
#define NH   256
#define NW   256
#define HW   (NH * NW)          // 65536

// ------------------------------ Phase 1 ------------------------------------
// 16*256*64 float4 outputs; each thread sums 68 channels for pred and gt.
__global__ void gp_channel_sum(const float* __restrict__ pred,
                               const float* __restrict__ gt,
                               float* __restrict__ P,
                               float* __restrict__ G) {
    int tid = blockIdx.x * blockDim.x + threadIdx.x;   // 0 .. 262143
    int w4  = tid & 63;
    int h   = (tid >> 6) & 255;
    int b   = tid >> 14;
    int pix = h * NW + (w4 << 2);
    long base = (long)b * NC * HW + pix;

    f4 accp = {0.f, 0.f, 0.f, 0.f};
    f4 accg = {0.f, 0.f, 0.f, 0.f};
    #pragma unroll 4
    for (int c = 0; c < NC; ++c) {
        accp += __builtin_nontemporal_load(
                    reinterpret_cast<const f4*>(pred + base + (long)c * HW));
        accg += __builtin_nontemporal_load(
                    reinterpret_cast<const f4*>(gt   + base + (long)c * HW));
    }
    int o = b * HW + pix;
    *reinterpret_cast<f4*>(P + o) = accp;   // RT store: stays in L2 for phase 2
    *reinterpret_cast<f4*>(G + o) = accg;
}

// ------------------------------ Phase 2 ------------------------------------
// grad(k) = P[off + k*sk] - P[off + k*sk + sd], zero-padded past L.
__device__ __forceinline__ float grad_at(const float* __restrict__ P,
                                         int off, int k, int sk, int sd, int L) {
    if (k >= L) return 0.f;
    int i = off + k * sk;
    return P[i] - P[i + sd];
}

__device__ __forceinline__ float diag_pick(v8f v, int r) {
    switch (r) {
        case 0: return v[0]; case 1: return v[1];
        case 2: return v[2]; case 3: return v[3];
        case 4: return v[4]; case 5: return v[5];
        case 6: return v[6]; default: return v[7];
    }
}

// One wave (32 lanes) per group of 16 vectors. 1022 groups total:
//   [0,256)    vertical rows    (L=255, n*B=4096)
//   [256,511)  vertical cols    (L=256, m*B=4080)
//   [511,766)  horizontal rows  (L=256, n*B=4080)
//   [766,1022) horizontal cols  (L=255, m*B=4096)
__global__ void gp_cosine(const float* __restrict__ P,
                          const float* __restrict__ G,
                          float* __restrict__ partial) {
    int g    = blockIdx.x;
    int lane = threadIdx.x;
    int m    = lane & 15;          // vector (row of A / col of B) this lane feeds

    int t, gi;
    if      (g < 256) { t = 0; gi = g;       }
    else if (g < 511) { t = 1; gi = g - 256; }
    else if (g < 766) { t = 2; gi = g - 511; }
    else              { t = 3; gi = g - 766; }
    int r = gi * 16 + m;

    int off, sk, sd, L;
    if (t == 0)      { int b = r >> 8,  h = r & 255; off = b*HW + h*NW; sk = 1;  sd = 1;  L = 255; }
    else if (t == 1) { int b = r / 255, w = r % 255; off = b*HW + w;    sk = NW; sd = 1;  L = 256; }
    else if (t == 2) { int b = r / 255, h = r % 255; off = b*HW + h*NW; sk = 1;  sd = NW; L = 256; }
    else             { int b = r >> 8,  w = r & 255; off = b*HW + w;    sk = NW; sd = NW; L = 255; }

    // A(16x4 f32) and B(4x16 f32) share the same per-lane mapping:
    // lanes 0-15: vector m=lane, K-sub {0,1}; lanes 16-31: vector m=lane-16, K-sub {2,3}.
    int ksub = (lane & 16) ? 2 : 0;

    v8f axy = {0.f,0.f,0.f,0.f,0.f,0.f,0.f,0.f};
    v8f axx = {0.f,0.f,0.f,0.f,0.f,0.f,0.f,0.f};
    v8f ayy = {0.f,0.f,0.f,0.f,0.f,0.f,0.f,0.f};

    for (int i = 0; i < 64; ++i) {         // K = 256 (zero-padded when L=255)
        int kb = i * 4 + ksub;
        float x0 = grad_at(P, off, kb,     sk, sd, L);
        float x1 = grad_at(P, off, kb + 1, sk, sd, L);
        float y0 = grad_at(G, off, kb,     sk, sd, L);
        float y1 = grad_at(G, off, kb + 1, sk, sd, L);
        v2f Ax = {x0, x1};
        v2f By = {y0, y1};
        // D = A x B + C ; diagonals accumulate Sum(x*y), Sum(x*x), Sum(y*y).
        axy = __builtin_amdgcn_wmma_f32_16x16x4_f32(false, Ax, false, By,
                                                    (short)0, axy, false, false);
        axx = __builtin_amdgcn_wmma_f32_16x16x4_f32(false, Ax, false, Ax,
                                                    (short)0, axx, false, false);
        ayy = __builtin_amdgcn_wmma_f32_16x16x4_f32(false, By, false, By,
                                                    (short)0, ayy, false, false);
    }

    // Diagonal (m,m): m<8 -> vgpr m, lane m ; m>=8 -> vgpr m-8, lane m+16.
    float c = 0.f;
    if (lane < 8 || lane >= 24) {
        int rr   = (lane < 8) ? lane : lane - 24;
        float xy = diag_pick(axy, rr);
        float xx = diag_pick(axx, rr);
        float yy = diag_pick(ayy, rr);
        c = xy / (fmaxf(sqrtf(xx), 1e-12f) * fmaxf(sqrtf(yy), 1e-12f));
    }
    // wave32 butterfly sum of the 16 cosines
    for (int o = 16; o > 0; o >>= 1)
        c += __shfl_xor(c, o, 32);
    if (lane == 0) partial[g] = c;
}

// ------------------------------ Phase 3 ------------------------------------
__global__ void gp_finalize(const float* __restrict__ partial,
                            float* __restrict__ out) {
    __shared__ float s[256];
    int tid = threadIdx.x;
    float acc = 0.f;
    for (int g = tid; g < 1022; g += 256) {
        float scale = (g < 256 || g >= 766) ? 4096.f : 4080.f;
        acc += partial[g] / scale;
    }
    s[tid] = acc;
    __syncthreads();
    for (int o = 128; o > 0; o >>= 1) {
        if (tid < o) s[tid] += s[tid + o];
        __syncthreads();
    }
    if (tid == 0) out[0] = -s[0];
}

// ---------------------------------------------------------------------------
extern "C" void kernel_launch(void* const* d_in, const int* in_sizes, int n_in,
                              void* d_out, int out_size, void* d_ws, size_t ws_size,
                              hipStream_t stream) {
    (void)in_sizes; (void)n_in; (void)out_size; (void)ws_size;
    const float* pred = (const float*)d_in[0];
    const float* gt   = (const float*)d_in[1];
    float* ws      = (float*)d_ws;
    float* P       = ws;                     // 16*65536 floats
    float* G       = ws + (long)NB * HW;     // 16*65536 floats
    float* partial = ws + 2L * NB * HW;      // 1022 floats

    gp_channel_sum<<<1024, 256, 0, stream>>>(pred, gt, P, G);
    gp_cosine<<<1022, 32, 0, stream>>>(P, G, partial);
    gp_finalize<<<1, 256, 0, stream>>>(partial, (float*)d_out);
}